// ReconstructionLoss_69741678953056
// MI455X (gfx1250) — compile-verified
//
#include <hip/hip_runtime.h>
#include <hip/hip_bf16.h>

// ReconstructionLoss for MI455X (gfx1250, wave32).
//
// Memory-bound: 268.4 MB read once, 4 B out -> HBM floor ~11.5us @ 23.3 TB/s.
// Strategy: streaming coalesced 128-bit non-temporal global loads (no reuse,
// so no LDS/TDM staging), per-thread f32 accumulation, and a full-precision
// wave-level reduction done with V_WMMA_F32_16X16X4_F32 (each lane's two f32
// partials are exactly the 2 A-matrix VGPRs of the 16x4 f32 A operand; B=ones
// gives row sums in D; 8 D-regs + one xor-16 shuffle = 32-lane wave sum).

typedef float v2f __attribute__((ext_vector_type(2)));
typedef float v4f __attribute__((ext_vector_type(4)));
typedef float v8f __attribute__((ext_vector_type(8)));

#define THREADS 256
#define NBLOCKS 4096

// Fixed problem geometry from the reference setup.
static constexpr int kF = 1024;          // frequency bins (magnitude half)
static constexpr int kLogF = 10;         // log2(kF)

__device__ __forceinline__ float wave_sum_via_wmma(float acc0, float acc1) {
    // A: 16x4 f32 striped across the wave (2 VGPRs/lane) = our 64 partials.
    // B: 4x16 all-ones. D[m][n] = sum_k A[m][k]. Every column of D is the
    // vector of row sums; lane n holds rows 0..7 (lanes 0-15) or 8..15
    // (lanes 16-31) in its 8 D registers.
    v2f A; A.x = acc0; A.y = acc1;
    v2f B; B.x = 1.0f; B.y = 1.0f;
    v8f C = {};
    v8f D = __builtin_amdgcn_wmma_f32_16x16x4_f32(
        /*neg_a=*/false, A, /*neg_b=*/false, B,
        /*c_mod=*/(short)0, C, /*reuse_a=*/false, /*reuse_b=*/false);
    float s = 0.0f;
#pragma unroll
    for (int i = 0; i < 8; ++i) s += D[i];
    // rows 0..7 half + rows 8..15 half -> full wave sum in every lane.
    s += __shfl_xor(s, 16, 32);
    return s;
}

__global__ __launch_bounds__(THREADS)
void recon_loss_partial(const float* __restrict__ masks,
                        const float* __restrict__ src_stft,
                        float* __restrict__ partial,
                        int nGroups) {
    const int tid    = blockIdx.x * blockDim.x + threadIdx.x;
    const int stride = gridDim.x * blockDim.x;

    const v4f* __restrict__ mv = (const v4f*)masks;
    const v4f* __restrict__ sv = (const v4f*)src_stft;

    float acc0 = 0.0f, acc1 = 0.0f;
    int toggle = 0;
    for (int g = tid; g < nGroups; g += stride) {
        // masks[g][0..3] : contiguous float4 (16B aligned)
        v4f m = __builtin_nontemporal_load(mv + g);
        // gt = src_stft[bt][f][0..3] with row length 2*F (magnitude half)
        int bt = g >> kLogF;
        int f  = g & (kF - 1);
        v4f gt = __builtin_nontemporal_load(sv + ((bt << (kLogF + 1)) + f));

        // first-max argmax over 4 channels (matches jnp.argmax tie rule)
        int idx = 0;
        float best = gt.x;
        if (gt.y > best) { best = gt.y; idx = 1; }
        if (gt.z > best) { best = gt.z; idx = 2; }
        if (gt.w > best) { best = gt.w; idx = 3; }

        float s = fabsf(m.x - (idx == 0 ? 1.0f : 0.0f))
                + fabsf(m.y - (idx == 1 ? 1.0f : 0.0f))
                + fabsf(m.z - (idx == 2 ? 1.0f : 0.0f))
                + fabsf(m.w - (idx == 3 ? 1.0f : 0.0f));
        if (toggle) acc1 += s; else acc0 += s;
        toggle ^= 1;
    }

    // Wave reduction on the matrix pipe (full f32 precision).
    float wsum = wave_sum_via_wmma(acc0, acc1);

    __shared__ float lsum[THREADS / 32];
    const int lane = threadIdx.x & 31;
    const int wave = threadIdx.x >> 5;
    if (lane == 0) lsum[wave] = wsum;
    __syncthreads();
    if (threadIdx.x == 0) {
        float b = 0.0f;
#pragma unroll
        for (int i = 0; i < THREADS / 32; ++i) b += lsum[i];
        partial[blockIdx.x] = b;
    }
}

__global__ __launch_bounds__(THREADS)
void recon_loss_final(const float* __restrict__ partial,
                      float* __restrict__ out,
                      int nPartials, float invCount) {
    float s = 0.0f;
    for (int i = threadIdx.x; i < nPartials; i += THREADS) s += partial[i];
    // Reuse the WMMA wave reduction here too (one per wave, negligible cost).
    float wsum = wave_sum_via_wmma(s, 0.0f);

    __shared__ float lsum[THREADS / 32];
    const int lane = threadIdx.x & 31;
    const int wave = threadIdx.x >> 5;
    if (lane == 0) lsum[wave] = wsum;
    __syncthreads();
    if (threadIdx.x == 0) {
        float t = 0.0f;
#pragma unroll
        for (int i = 0; i < THREADS / 32; ++i) t += lsum[i];
        out[0] = t * invCount;   // invCount = 2^-25 -> exact scaling
    }
}

extern "C" void kernel_launch(void* const* d_in, const int* in_sizes, int n_in,
                              void* d_out, int out_size, void* d_ws, size_t ws_size,
                              hipStream_t stream) {
    const float* masks    = (const float*)d_in[0];  // [B,T,F,C] f32
    const float* src_stft = (const float*)d_in[1];  // [B,T,2F,C] f32
    float* out = (float*)d_out;                     // scalar f32

    const int nMasks  = in_sizes[0];                // B*T*F*C = 33,554,432
    const int nGroups = nMasks / 4;                 // B*T*F   =  8,388,608
    float* partial = (float*)d_ws;                  // NBLOCKS floats (16 KB)

    recon_loss_partial<<<NBLOCKS, THREADS, 0, stream>>>(masks, src_stft, partial, nGroups);

    const float invCount = 1.0f / (float)nMasks;    // 1/2^25, exact
    recon_loss_final<<<1, THREADS, 0, stream>>>(partial, out, NBLOCKS, invCount);
}